// softIoULoss_7000796692664
// MI455X (gfx1250) — compile-verified
//
#include <hip/hip_runtime.h>

typedef float v2f __attribute__((ext_vector_type(2)));
typedef float v4f __attribute__((ext_vector_type(4)));
typedef float v8f __attribute__((ext_vector_type(8)));

#define SIOU_EPS 0.5f

// Exact full-wave (32-lane) f32 sum using the f32 WMMA pipe:
//   A = ones(16x4), B[k][n] built so column n = v_n + v_{n+16}; then 4 xor-shuffles
// All adds are plain f32 adds inside the matrix unit -> same arithmetic as shuffles.
// EXEC must be all ones (call only from reconverged, full-block code).
__device__ __forceinline__ float wave_sum32(float v) {
    v2f a; a[0] = 1.0f; a[1] = 1.0f;   // A-matrix: all ones
    v2f b; b[0] = v;    b[1] = 0.0f;   // B-matrix: lane value in rows K0/K1, zeros in K2/K3
    v8f c = {};
    c = __builtin_amdgcn_wmma_f32_16x16x4_f32(
            /*neg_a=*/false, a, /*neg_b=*/false, b,
            /*c_mod=*/(short)0, c, /*reuse_a=*/false, /*reuse_b=*/false);
    float p = c[0];                    // lane L holds v_(L%16) + v_(L%16 + 16)
    p += __shfl_xor(p, 1, 32);
    p += __shfl_xor(p, 2, 32);
    p += __shfl_xor(p, 4, 32);
    p += __shfl_xor(p, 8, 32);
    return p;                          // full 32-lane sum, replicated in every lane
}

__global__ void __launch_bounds__(256)
siou_partial_kernel(const float* __restrict__ x, const float* __restrict__ t,
                    const int* __restrict__ ncp, float* __restrict__ part,
                    long long n) {
    const float ncf = (float)(*ncp);
    const long long tid    = (long long)blockIdx.x * blockDim.x + threadIdx.x;
    const long long stride = (long long)gridDim.x * blockDim.x;
    const long long nv     = n >> 2;

    const v4f* __restrict__ x4 = (const v4f*)x;
    const v4f* __restrict__ t4 = (const v4f*)t;

    float s_invs = 0.0f, s_match = 0.0f, s_inter = 0.0f;

    // streaming (non-temporal) B128 loads: data is read exactly once
    for (long long i = tid; i < nv; i += stride) {
        v4f xv = __builtin_nontemporal_load(&x4[i]);
        v4f tv = __builtin_nontemporal_load(&t4[i]);
#pragma unroll
        for (int c = 0; c < 4; ++c) {
            float xs = xv[c];
            float ts = tv[c];
            float d   = __builtin_fabsf(xs - ts);
            float inv = __builtin_amdgcn_rcpf(1.0f + __expf(d - SIOU_EPS)); // sigmoid(eps-|x-t|)
            float m   = ((ts == __builtin_floorf(ts)) && (ts >= 0.0f) && (ts <= ncf))
                            ? 1.0f : 0.0f;
            s_invs  += inv;
            s_match += m;
            s_inter += inv * m;
        }
    }
    // scalar tail (n % 4), zero-trip for the reference sizes
    for (long long i = (nv << 2) + tid; i < n; i += stride) {
        float xs = x[i], ts = t[i];
        float d   = __builtin_fabsf(xs - ts);
        float inv = __builtin_amdgcn_rcpf(1.0f + __expf(d - SIOU_EPS));
        float m   = ((ts == __builtin_floorf(ts)) && (ts >= 0.0f) && (ts <= ncf))
                        ? 1.0f : 0.0f;
        s_invs += inv; s_match += m; s_inter += inv * m;
    }

    // ---- wave reduction via f32 WMMA (all lanes reconverged here) ----
    float w_invs  = wave_sum32(s_invs);
    float w_match = wave_sum32(s_match);
    float w_inter = wave_sum32(s_inter);

    __shared__ float red[8 * 3];   // 256 threads -> 8 wave32 per block
    const int wave = threadIdx.x >> 5;
    if ((threadIdx.x & 31) == 0) {
        red[wave * 3 + 0] = w_invs;
        red[wave * 3 + 1] = w_match;
        red[wave * 3 + 2] = w_inter;
    }
    __syncthreads();
    if (threadIdx.x == 0) {
        float a = 0.0f, b = 0.0f, c = 0.0f;
#pragma unroll
        for (int w = 0; w < 8; ++w) {
            a += red[w * 3 + 0];
            b += red[w * 3 + 1];
            c += red[w * 3 + 2];
        }
        part[(long long)blockIdx.x * 3 + 0] = a;  // sum invsigm
        part[(long long)blockIdx.x * 3 + 1] = b;  // sum matched
        part[(long long)blockIdx.x * 3 + 2] = c;  // intersection
    }
}

__global__ void __launch_bounds__(256)
siou_finalize_kernel(const float* __restrict__ part, int nblocks,
                     const int* __restrict__ ncp, float* __restrict__ out) {
    double a = 0.0, b = 0.0, c = 0.0;
    for (int i = threadIdx.x; i < nblocks; i += blockDim.x) {
        a += (double)part[i * 3 + 0];
        b += (double)part[i * 3 + 1];
        c += (double)part[i * 3 + 2];
    }
    // 32-lane reduction on doubles
    for (int m = 16; m > 0; m >>= 1) {
        a += __shfl_xor(a, m, 32);
        b += __shfl_xor(b, m, 32);
        c += __shfl_xor(c, m, 32);
    }
    __shared__ double red[8 * 3];
    const int wave = threadIdx.x >> 5;
    if ((threadIdx.x & 31) == 0) {
        red[wave * 3 + 0] = a;
        red[wave * 3 + 1] = b;
        red[wave * 3 + 2] = c;
    }
    __syncthreads();
    if (threadIdx.x == 0) {
        double s_invs = 0.0, s_match = 0.0, s_inter = 0.0;
#pragma unroll
        for (int w = 0; w < 8; ++w) {
            s_invs  += red[w * 3 + 0];
            s_match += red[w * 3 + 1];
            s_inter += red[w * 3 + 2];
        }
        const double ncp1   = (double)(*ncp + 1);
        const double smooth = 1.0;
        const double uni    = s_match + ncp1 * s_invs - s_inter;
        const double sIoU   = (s_inter + smooth) / (uni + smooth);
        out[0] = (float)(1.0 - sIoU);
    }
}

extern "C" void kernel_launch(void* const* d_in, const int* in_sizes, int n_in,
                              void* d_out, int out_size, void* d_ws, size_t ws_size,
                              hipStream_t stream) {
    const float* x  = (const float*)d_in[0];
    const float* t  = (const float*)d_in[1];
    const int* ncp  = (const int*)d_in[2];
    float* out      = (float*)d_out;
    float* part     = (float*)d_ws;

    const long long n = (long long)in_sizes[0];

    int blocks = 4096;                                   // 48 KB of ws partials
    const int maxBlocksWs = (int)(ws_size / (3 * sizeof(float)));
    if (blocks > maxBlocksWs) blocks = maxBlocksWs;
    if (blocks < 1) blocks = 1;

    siou_partial_kernel<<<blocks, 256, 0, stream>>>(x, t, ncp, part, n);
    siou_finalize_kernel<<<1, 256, 0, stream>>>(part, blocks, ncp, out);
}